// DeepGCN_30039001268347
// MI455X (gfx1250) — compile-verified
//
#include <hip/hip_runtime.h>
#include <hip/hip_bf16.h>
#include <math.h>

// ---------------- constants (match reference) ----------------
#define NN   50000
#define EE   1200000
#define GG   64
#define HH   64
#define HID  128
#define FN   128
#define FE   32
#define LL   3

typedef float v2f __attribute__((ext_vector_type(2)));
typedef float v8f __attribute__((ext_vector_type(8)));

// order-preserving float<->int for atomic max on floats
__device__ __forceinline__ int encf(float f) {
    int i = __float_as_int(f);
    return i >= 0 ? i : (i ^ 0x7FFFFFFF);
}
__device__ __forceinline__ float decf(int k) {
    return __int_as_float(k >= 0 ? k : (k ^ 0x7FFFFFFF));
}

// ---------------------------------------------------------------------------
// Generic fp32 WMMA GEMM:  C[M,Nout] = A[M,K] @ W[Nout,K]^T + bias (+R) (relu)
// One wave computes one 16x16 tile via V_WMMA_F32_16X16X4_F32, K stepped by 4.
// ---------------------------------------------------------------------------
__global__ __launch_bounds__(256) void wmma_gemm_nt(
    const float* __restrict__ A, const float* __restrict__ W,
    const float* __restrict__ bias, const float* __restrict__ R,
    float* __restrict__ C, int M, int K, int Nout, int do_relu) {
    const int lane = threadIdx.x & 31;
    const int wave = threadIdx.x >> 5;
    const int rt = blockIdx.x * 8 + wave;           // 16-row tile
    if (rt * 16 >= M) return;                        // wave-uniform exit
    const int ct = blockIdx.y;                       // 16-col tile
    const int mr = lane & 15;
    const int kh = lane >> 4;                        // K-half select
    const int col = ct * 16 + mr;
    const float* arow = A + (size_t)(rt * 16 + mr) * K + 2 * kh;
    const float* brow = W + (size_t)col * K + 2 * kh;
    v8f acc = {};
    for (int k0 = 0; k0 < K; k0 += 4) {
        v2f a = *(const v2f*)(arow + k0);
        v2f b = *(const v2f*)(brow + k0);
        acc = __builtin_amdgcn_wmma_f32_16x16x4_f32(false, a, false, b,
                                                    (short)0, acc, false, false);
    }
    float bc = bias[col];
#pragma unroll
    for (int i = 0; i < 8; ++i) {
        int row = rt * 16 + 8 * kh + i;
        float v = acc[i] + bc;
        if (do_relu) v = fmaxf(v, 0.f);
        size_t idx = (size_t)row * Nout + col;
        if (R) v += R[idx];
        C[idx] = v;
    }
}

// ---------------------------------------------------------------------------
// Fused edge encoder: ea = relu(EA @ W1^T + b1) @ W2^T + b2   [E,32]->[E,64]
// Stage-1 result staged through per-wave LDS tile (C-layout -> A-layout).
// ---------------------------------------------------------------------------
__global__ __launch_bounds__(256) void edge_encoder_kernel(
    const float* __restrict__ EA, const float* __restrict__ W1,
    const float* __restrict__ B1, const float* __restrict__ W2,
    const float* __restrict__ B2, float* __restrict__ ea_out, int E) {
    __shared__ float T[8][16][32];
    const int lane = threadIdx.x & 31;
    const int wave = threadIdx.x >> 5;
    const int et = blockIdx.x * 8 + wave;
    if (et * 16 >= E) return;
    const int mr = lane & 15;
    const int kh = lane >> 4;
    const float* arow = EA + (size_t)(et * 16 + mr) * FE + 2 * kh;
    // stage 1: [16,32] = relu(EA_tile @ W1^T + b1)
#pragma unroll
    for (int ct = 0; ct < 2; ++ct) {
        const int col = ct * 16 + mr;
        const float* brow = W1 + col * FE + 2 * kh;
        v8f acc = {};
#pragma unroll
        for (int k0 = 0; k0 < FE; k0 += 4) {
            v2f a = *(const v2f*)(arow + k0);
            v2f b = *(const v2f*)(brow + k0);
            acc = __builtin_amdgcn_wmma_f32_16x16x4_f32(false, a, false, b,
                                                        (short)0, acc, false, false);
        }
        float bc = B1[col];
#pragma unroll
        for (int i = 0; i < 8; ++i)
            T[wave][8 * kh + i][col] = fmaxf(acc[i] + bc, 0.f);
    }
    // stage 2: ea[16,64] = T @ W2^T + b2 (A-frags re-read from LDS; same-wave
    // DS dependency, compiler inserts s_wait_dscnt)
#pragma unroll
    for (int ct = 0; ct < 4; ++ct) {
        const int col = ct * 16 + mr;
        const float* brow = W2 + col * FE + 2 * kh;
        v8f acc = {};
#pragma unroll
        for (int k0 = 0; k0 < FE; k0 += 4) {
            v2f a;
            a.x = T[wave][mr][k0 + 2 * kh];
            a.y = T[wave][mr][k0 + 2 * kh + 1];
            v2f b = *(const v2f*)(brow + k0);
            acc = __builtin_amdgcn_wmma_f32_16x16x4_f32(false, a, false, b,
                                                        (short)0, acc, false, false);
        }
        float bc = B2[col];
#pragma unroll
        for (int i = 0; i < 8; ++i)
            ea_out[(size_t)(et * 16 + 8 * kh + i) * HH + col] = acc[i] + bc;
    }
}

// ---------------- per-layer node-array init ----------------
__global__ void init_nh_kernel(int* __restrict__ m, float* __restrict__ den,
                               float* __restrict__ num, int n) {
    int i = blockIdx.x * blockDim.x + threadIdx.x;
    if (i < n) { m[i] = encf(-INFINITY); den[i] = 0.f; num[i] = 0.f; }
}

// ---------------- edge pass 1: segment max of s = (relu(z[src]+ea)+eps)*t ----
__global__ __launch_bounds__(256) void edge_max_kernel(
    const float* __restrict__ zin, const float* __restrict__ ea,
    const int* __restrict__ src, const int* __restrict__ dst,
    const float* __restrict__ tptr, int* __restrict__ menc, int E) {
    int gid = blockIdx.x * 256 + threadIdx.x;
    int e = gid >> 4;
    if (e >= E) return;
    int q = (gid & 15) * 4;
    if (e + 256 < E) __builtin_prefetch(ea + (size_t)(e + 256) * HH + q, 0, 1);
    float t = *tptr;
    int s = src[e], d = dst[e];
    float4 hz = *(const float4*)(zin + (size_t)s * HH + q);
    float4 av = *(const float4*)(ea + (size_t)e * HH + q);
    float m0 = fmaxf(hz.x + av.x, 0.f) + 1e-7f;
    float m1 = fmaxf(hz.y + av.y, 0.f) + 1e-7f;
    float m2 = fmaxf(hz.z + av.z, 0.f) + 1e-7f;
    float m3 = fmaxf(hz.w + av.w, 0.f) + 1e-7f;
    int* mp = menc + (size_t)d * HH + q;
    atomicMax(mp + 0, encf(m0 * t));
    atomicMax(mp + 1, encf(m1 * t));
    atomicMax(mp + 2, encf(m2 * t));
    atomicMax(mp + 3, encf(m3 * t));
}

// ---------------- edge pass 2: num += msg*exp(s-m), den += exp(s-m) ----------
__global__ __launch_bounds__(256) void edge_sum_kernel(
    const float* __restrict__ zin, const float* __restrict__ ea,
    const int* __restrict__ src, const int* __restrict__ dst,
    const float* __restrict__ tptr, const int* __restrict__ menc,
    float* __restrict__ den, float* __restrict__ num, int E) {
    int gid = blockIdx.x * 256 + threadIdx.x;
    int e = gid >> 4;
    if (e >= E) return;
    int q = (gid & 15) * 4;
    if (e + 256 < E) __builtin_prefetch(ea + (size_t)(e + 256) * HH + q, 0, 1);
    float t = *tptr;
    int s = src[e], d = dst[e];
    float4 hz = *(const float4*)(zin + (size_t)s * HH + q);
    float4 av = *(const float4*)(ea + (size_t)e * HH + q);
    size_t base = (size_t)d * HH + q;
    float msg[4] = { fmaxf(hz.x + av.x, 0.f) + 1e-7f, fmaxf(hz.y + av.y, 0.f) + 1e-7f,
                     fmaxf(hz.z + av.z, 0.f) + 1e-7f, fmaxf(hz.w + av.w, 0.f) + 1e-7f };
#pragma unroll
    for (int j = 0; j < 4; ++j) {
        float mval = decf(menc[base + j]);
        float a = __expf(msg[j] * t - mval);
        atomicAdd(den + base + j, a);
        atomicAdd(num + base + j, msg[j] * a);
    }
}

// ---------------- combine: outn = num/(den+1e-16) + conv_input --------------
__global__ void combine_kernel(const float* __restrict__ num,
                               const float* __restrict__ den,
                               const float* __restrict__ cin,
                               float* __restrict__ outn, int n) {
    int i = blockIdx.x * blockDim.x + threadIdx.x;
    if (i < n) outn[i] = num[i] / (den[i] + 1e-16f) + cin[i];
}

// ---------------- LayerNorm(+ReLU), width 128, in-place (wave per row) ------
__global__ __launch_bounds__(256) void ln_relu_128_kernel(
    float* __restrict__ buf, const float* __restrict__ g,
    const float* __restrict__ b, int M) {
    int row = blockIdx.x * 8 + (threadIdx.x >> 5);
    if (row >= M) return;
    int lane = threadIdx.x & 31;
    int c = lane * 4;
    float4 v = *(const float4*)(buf + (size_t)row * HID + c);
    float s = v.x + v.y + v.z + v.w;
    for (int m = 16; m >= 1; m >>= 1) s += __shfl_xor(s, m, 32);
    float mu = s * (1.f / HID);
    float dx = v.x - mu, dy = v.y - mu, dz = v.z - mu, dw = v.w - mu;
    float q = dx * dx + dy * dy + dz * dz + dw * dw;
    for (int m = 16; m >= 1; m >>= 1) q += __shfl_xor(q, m, 32);
    float inv = rsqrtf(q * (1.f / HID) + 1e-5f);
    float4 o;
    o.x = fmaxf(dx * inv * g[c + 0] + b[c + 0], 0.f);
    o.y = fmaxf(dy * inv * g[c + 1] + b[c + 1], 0.f);
    o.z = fmaxf(dz * inv * g[c + 2] + b[c + 2], 0.f);
    o.w = fmaxf(dw * inv * g[c + 3] + b[c + 3], 0.f);
    *(float4*)(buf + (size_t)row * HID + c) = o;
}

// ---------------- LayerNorm(+ReLU), width 64, h -> zout (wave per row) ------
__global__ __launch_bounds__(256) void ln_relu_64_kernel(
    const float* __restrict__ hin, const float* __restrict__ g,
    const float* __restrict__ b, float* __restrict__ zout, int M) {
    int row = blockIdx.x * 8 + (threadIdx.x >> 5);
    if (row >= M) return;
    int lane = threadIdx.x & 31;
    int c = lane * 2;
    float2 v = *(const float2*)(hin + (size_t)row * HH + c);
    float s = v.x + v.y;
    for (int m = 16; m >= 1; m >>= 1) s += __shfl_xor(s, m, 32);
    float mu = s * (1.f / HH);
    float dx = v.x - mu, dy = v.y - mu;
    float q = dx * dx + dy * dy;
    for (int m = 16; m >= 1; m >>= 1) q += __shfl_xor(q, m, 32);
    float inv = rsqrtf(q * (1.f / HH) + 1e-5f);
    float2 o;
    o.x = fmaxf(dx * inv * g[c + 0] + b[c + 0], 0.f);
    o.y = fmaxf(dy * inv * g[c + 1] + b[c + 1], 0.f);
    *(float2*)(zout + (size_t)row * HH + c) = o;
}

// ---------------- global max pool over batch --------------------------------
__global__ void init_pool_kernel(int* __restrict__ p, int n) {
    int i = blockIdx.x * blockDim.x + threadIdx.x;
    if (i < n) p[i] = encf(-INFINITY);
}
__global__ void pool_kernel(const float* __restrict__ z, const int* __restrict__ batch,
                            int* __restrict__ pooled, int N) {
    int gid = blockIdx.x * blockDim.x + threadIdx.x;
    int node = gid >> 6;
    if (node >= N) return;
    int c = gid & 63;
    int g = batch[node];
    atomicMax(pooled + (size_t)g * HH + c, encf(z[(size_t)node * HH + c]));
}

// ---------------- readout: sigmoid(pooled @ ro_w^T + ro_b) ------------------
__global__ void readout_kernel(const int* __restrict__ pooled,
                               const float* __restrict__ w,
                               const float* __restrict__ b,
                               float* __restrict__ out, int G) {
    int g = blockIdx.x * blockDim.x + threadIdx.x;
    if (g >= G) return;
    float acc = b[0];
    for (int c = 0; c < HH; ++c) acc += decf(pooled[(size_t)g * HH + c]) * w[c];
    out[g] = 1.f / (1.f + __expf(-acc));
}

// ---------------------------------------------------------------------------
extern "C" void kernel_launch(void* const* d_in, const int* in_sizes, int n_in,
                              void* d_out, int out_size, void* d_ws, size_t ws_size,
                              hipStream_t stream) {
    const float* x        = (const float*)d_in[0];
    const float* eattr    = (const float*)d_in[1];
    const float* ne_w     = (const float*)d_in[2];
    const float* ne_b     = (const float*)d_in[3];
    const float* ee_w1    = (const float*)d_in[4];
    const float* ee_b1    = (const float*)d_in[5];
    const float* ee_w2    = (const float*)d_in[6];
    const float* ee_b2    = (const float*)d_in[7];
    const float* conv_w1  = (const float*)d_in[8];
    const float* conv_b1  = (const float*)d_in[9];
    const float* conv_g   = (const float*)d_in[10];
    const float* conv_bn  = (const float*)d_in[11];
    const float* conv_w2  = (const float*)d_in[12];
    const float* conv_b2  = (const float*)d_in[13];
    const float* conv_t   = (const float*)d_in[14];
    const float* ln_g     = (const float*)d_in[15];
    const float* ln_b     = (const float*)d_in[16];
    const float* ro_w     = (const float*)d_in[17];
    const float* ro_b     = (const float*)d_in[18];
    const int*   ei       = (const int*)d_in[19];
    const int*   batch    = (const int*)d_in[20];
    const int* src = ei;
    const int* dst = ei + EE;
    float* out = (float*)d_out;

    // workspace carve (all 16B aligned: every slab is a multiple of 16 bytes)
    char* w = (char*)d_ws;
    float* h    = (float*)w; w += (size_t)NN * HH * 4;
    float* z    = (float*)w; w += (size_t)NN * HH * 4;
    float* outn = (float*)w; w += (size_t)NN * HH * 4;
    float* den  = (float*)w; w += (size_t)NN * HH * 4;
    float* num  = (float*)w; w += (size_t)NN * HH * 4;
    int*   menc = (int*)w;   w += (size_t)NN * HH * 4;
    float* mid  = (float*)w; w += (size_t)NN * HID * 4;
    int*   pooled = (int*)w; w += (size_t)GG * HH * 4;
    float* ea   = (float*)w; w += (size_t)EE * HH * 4;

    const int gemm_rows_N = (NN / 16 + 7) / 8;       // 391 blocks of 8 waves
    const dim3 blk(256);

    // node encoder: h = x @ ne_w^T + ne_b   [N,128]->[N,64]
    wmma_gemm_nt<<<dim3(gemm_rows_N, HH / 16), blk, 0, stream>>>(
        x, ne_w, ne_b, nullptr, h, NN, FN, HH, 0);

    // edge encoder (fused two GEMMs): ea [E,64]
    edge_encoder_kernel<<<dim3(EE / 16 / 8), blk, 0, stream>>>(
        eattr, ee_w1, ee_b1, ee_w2, ee_b2, ea, EE);

    const int nNH   = NN * HH;
    const int bNH   = (nNH + 255) / 256;
    const int bEdge = (EE * 16 + 255) / 256;

    for (int l = 0; l < LL; ++l) {
        const float* cin = h;
        if (l > 0) {
            ln_relu_64_kernel<<<dim3((NN + 7) / 8), blk, 0, stream>>>(
                h, ln_g + l * HH, ln_b + l * HH, z, NN);
            cin = z;
        }
        init_nh_kernel<<<dim3(bNH), blk, 0, stream>>>(menc, den, num, nNH);
        edge_max_kernel<<<dim3(bEdge), blk, 0, stream>>>(
            cin, ea, src, dst, conv_t + l, menc, EE);
        edge_sum_kernel<<<dim3(bEdge), blk, 0, stream>>>(
            cin, ea, src, dst, conv_t + l, menc, den, num, EE);
        combine_kernel<<<dim3(bNH), blk, 0, stream>>>(num, den, cin, outn, nNH);
        // MLP: mid = outn @ w1^T + b1 ; LN+ReLU ; h (+)= mid @ w2^T + b2
        wmma_gemm_nt<<<dim3(gemm_rows_N, HID / 16), blk, 0, stream>>>(
            outn, conv_w1 + (size_t)l * HID * HH, conv_b1 + l * HID,
            nullptr, mid, NN, HH, HID, 0);
        ln_relu_128_kernel<<<dim3((NN + 7) / 8), blk, 0, stream>>>(
            mid, conv_g + l * HID, conv_bn + l * HID, NN);
        wmma_gemm_nt<<<dim3(gemm_rows_N, HH / 16), blk, 0, stream>>>(
            mid, conv_w2 + (size_t)l * HH * HID, conv_b2 + l * HH,
            (l == 0 ? nullptr : h), h, NN, HID, HH, 0);
    }

    // final act-norm with layer-0 params, then global max pool + readout
    ln_relu_64_kernel<<<dim3((NN + 7) / 8), blk, 0, stream>>>(
        h, ln_g, ln_b, z, NN);
    init_pool_kernel<<<dim3((GG * HH + 255) / 256), blk, 0, stream>>>(pooled, GG * HH);
    pool_kernel<<<dim3(bNH), blk, 0, stream>>>(z, batch, pooled, NN);
    readout_kernel<<<dim3(1), dim3(64), 0, stream>>>(pooled, ro_w, ro_b, out, GG);
}